// LatticeSuperpixel_66417374265609
// MI455X (gfx1250) — compile-verified
//
#include <hip/hip_runtime.h>
#include <hip/hip_bf16.h>

typedef float v2f __attribute__((ext_vector_type(2)));
typedef float v8f __attribute__((ext_vector_type(8)));
typedef unsigned int u32x4 __attribute__((ext_vector_type(4)));
typedef int i32x8 __attribute__((ext_vector_type(8)));
typedef int i32x4 __attribute__((ext_vector_type(4)));

#define BATCH 8
#define CHN   20
#define HH    512
#define WW    512
#define SH    32
#define SW    32
#define SEG   (SH*SW)      // 1024 seeds
#define CELL  16           // 16x16 pixels per seed cell
#define PIX   256
#define XSTR  257          // LDS stride pad (257 % 64 == 1 -> conflict-free)
#define EPSV  1e-8f

__device__ __forceinline__ int clampi(int v, int lo, int hi) {
    return v < lo ? lo : (v > hi ? hi : v);
}

// ---------------------------------------------------------------------------
// seed_init: seeds[b][s][c] = mean over 16x16 cell; ss[b][s] = sum_c seed^2
// one block (256 thr) per (b, s); float4 global loads, LDS f32 atomics
// ---------------------------------------------------------------------------
__global__ __launch_bounds__(256)
void seed_init_kernel(const float* __restrict__ x,
                      float* __restrict__ seeds,
                      float* __restrict__ ss)
{
    __shared__ float accs[CHN];
    const int blk = blockIdx.x;
    const int b  = blk >> 10;
    const int s  = blk & (SEG - 1);
    const int sy = s >> 5, sx = s & 31;
    const int t  = threadIdx.x;

    if (t < CHN) accs[t] = 0.f;
    __syncthreads();

    const int q     = t & 63;        // pixel quad index (row*4 + quad)
    const int cbase = t >> 6;        // 0..3
    const int py = sy * CELL + (q >> 2);
    const int px = sx * CELL + (q & 3) * 4;

    for (int c = cbase; c < CHN; c += 4) {
        const float4 v = *(const float4*)(x + (size_t)(b * CHN + c) * HH * WW
                                            + (size_t)py * WW + px);
        atomicAdd(&accs[c], v.x + v.y + v.z + v.w);
    }
    __syncthreads();

    if (t < CHN) seeds[(size_t)(b * SEG + s) * CHN + t] = accs[t] * (1.f / 256.f);
    __syncthreads();
    if (t == 0) {
        float sq = 0.f;
        #pragma unroll
        for (int c = 0; c < CHN; ++c) {
            float v = accs[c] * (1.f / 256.f);
            sq += v * v;
        }
        ss[b * SEG + s] = sq;
    }
}

// ---------------------------------------------------------------------------
// zero accumulators (numer || denom contiguous)
// ---------------------------------------------------------------------------
__global__ void zero_kernel(float* __restrict__ p, int n)
{
    int i = blockIdx.x * blockDim.x + threadIdx.x;
    if (i < n) p[i] = 0.f;
}

// ---------------------------------------------------------------------------
// cell_kernel: one block per (b, cell).
//   TDM tensor_load_to_lds (3D tile 16x16x20, HW padding -> stride 257)
//   -> dist via V_WMMA_F32_16X16X4_F32 (B pre-scaled by -2, C = |s|^2)
//   -> per-thread softmax over 9 offsets (no cross-lane ops)
//   -> Q to d_out (last iter) or numer/denom accumulation (other iters)
// ---------------------------------------------------------------------------
__global__ __launch_bounds__(256)
void cell_kernel(const float* __restrict__ x,
                 const float* __restrict__ seeds,
                 const float* __restrict__ ss,
                 float* __restrict__ numer,
                 float* __restrict__ denom,
                 float* __restrict__ qout,
                 int last)
{
    __shared__ float x_lds[CHN * XSTR];   // [c][p], padded (filled by TDM)
    __shared__ float dq_lds[9 * XSTR];    // dist then Q, [o][p], padded
    __shared__ float s_lds[CHN * 16];     // -2 * seed, [c][o], o padded to 16
    __shared__ float ss_lds[16];          // |s_o|^2
    __shared__ int   sidx[9];             // neighbor seed indices

    const int blk = blockIdx.x;
    const int b  = blk >> 10;
    const int s  = blk & (SEG - 1);
    const int sy = s >> 5, sx = s & 31;
    const int t    = threadIdx.x;
    const int lane = t & 31;
    const int wave = t >> 5;
    const int oN   = lane & 15;
    const int half = lane >> 4;

    if (t < 9) {
        int dy = t / 3 - 1, dx = t % 3 - 1;
        int iy = clampi(sy + dy, 0, SH - 1);
        int ix = clampi(sx + dx, 0, SW - 1);
        sidx[t] = iy * SW + ix;
    }
    if (t < 16) ss_lds[t] = 0.f;

    // Kick off the TDM load of the x tile: 3D tile X=16, Y=16, Z=20 channels,
    // row stride W, channel stride H*W; LDS padding 1 dword per 256 dwords
    // reproduces x_lds[c*257 + p].
    if (wave == 0) {
        const unsigned long long ga =
            (unsigned long long)(uintptr_t)(x + (size_t)(b * CHN) * HH * WW
                                              + (size_t)(sy * CELL) * WW + sx * CELL);
        const unsigned lds_off = (unsigned)(uintptr_t)(&x_lds[0]);
        u32x4 g0;
        g0[0] = 1u;                                   // count=1 (valid descriptor)
        g0[1] = lds_off;                              // lds_addr
        g0[2] = (unsigned)ga;                         // global_addr[31:0]
        g0[3] = ((unsigned)(ga >> 32) & 0x01FFFFFFu)  // global_addr[56:32]
                | 0x80000000u;                        // type=2 ("image")
        i32x8 g1;
        g1[0] = 0x01D20000;          // data_size=4B, pad_enable, intv=256dw, amt=1dw
        g1[1] = (int)(512u << 16);   // tensor_dim0 = 512 (low16)
        g1[2] = (int)(512u << 16);   // tensor_dim0 hi=0 | tensor_dim1 = 512 (low16)
        g1[3] = (int)(16u << 16);    // tensor_dim1 hi=0 | tile_dim0 = 16
        g1[4] = (int)((20u << 16) | 16u);  // tile_dim1=16 | tile_dim2=20
        g1[5] = 512;                 // tensor_dim0_stride (row stride)
        g1[6] = 0;                   // stride0 hi | stride1 low16 (0x40000 -> 0)
        g1[7] = 4;                   // tensor_dim1_stride[47:16] = 0x40000>>16
        i32x4 g2;
        g2[0] = CHN;                 // tensor_dim2 = 20
        g2[1] = 0; g2[2] = 0; g2[3] = 0;
        i32x4 g3; g3[0] = 0; g3[1] = 0; g3[2] = 0; g3[3] = 0;
        i32x8 g4;
        g4[0] = 0; g4[1] = 0; g4[2] = 0; g4[3] = 0;
        g4[4] = 0; g4[5] = 0; g4[6] = 0; g4[7] = 0;
        __builtin_amdgcn_tensor_load_to_lds(g0, g1, g2, g3, g4, 0);
    }
    __syncthreads();   // sidx visible

    // seeds for the 9 candidates, pre-scaled by -2, padded to 16 cols with 0
    for (int i = t; i < CHN * 16; i += 256) {
        int c = i >> 4, o = i & 15;
        s_lds[i] = (o < 9) ? -2.f * seeds[(size_t)(b * SEG + sidx[o]) * CHN + c] : 0.f;
    }
    if (t < 9) ss_lds[t] = ss[b * SEG + sidx[t]];

    if (wave == 0) __builtin_amdgcn_s_wait_tensorcnt(0);
    __syncthreads();   // x_lds (TDM), s_lds, ss_lds all visible

    // dist_rel[p][o] = |s_o|^2 - 2 x_p . s_o  (the |x_p|^2 term cancels in softmax)
    // each wave handles pixel tiles 2*wave, 2*wave+1
    #pragma unroll
    for (int ti = 0; ti < 2; ++ti) {
        const int T = wave * 2 + ti;
        const float cinit = ss_lds[oN];
        v8f acc = {cinit, cinit, cinit, cinit, cinit, cinit, cinit, cinit};
        #pragma unroll
        for (int kb = 0; kb < CHN; kb += 4) {
            const int k0 = kb + 2 * half;
            v2f a, bf;
            a.x  = x_lds[(k0 + 0) * XSTR + T * 16 + oN];
            a.y  = x_lds[(k0 + 1) * XSTR + T * 16 + oN];
            bf.x = s_lds[(k0 + 0) * 16 + oN];
            bf.y = s_lds[(k0 + 1) * 16 + oN];
            acc = __builtin_amdgcn_wmma_f32_16x16x4_f32(
                false, a, false, bf, (short)0, acc, false, false);
        }
        if (oN < 9) {
            #pragma unroll
            for (int v = 0; v < 8; ++v)
                dq_lds[oN * XSTR + T * 16 + v + 8 * half] = acc[v];
        }
    }
    __syncthreads();

    // per-thread softmax over the 9 offsets for pixel p = t (no cross-lane ops)
    {
        float d[9], e[9];
        float dmin = 3.4e38f;
        #pragma unroll
        for (int o = 0; o < 9; ++o) {
            d[o] = dq_lds[o * XSTR + t];
            dmin = fminf(dmin, d[o]);
        }
        float esum = 0.f;
        #pragma unroll
        for (int o = 0; o < 9; ++o) {
            e[o] = __expf(dmin - d[o]);
            esum += e[o];
        }
        const float inv = 1.f / esum;
        if (last) {
            const int py = sy * CELL + (t >> 4);
            const int px = sx * CELL + (t & 15);
            #pragma unroll
            for (int o = 0; o < 9; ++o)
                qout[(((size_t)b * 9 + o) * HH + py) * WW + px] = e[o] * inv;
        } else {
            #pragma unroll
            for (int o = 0; o < 9; ++o)
                dq_lds[o * XSTR + t] = e[o] * inv;   // in place: own column only
        }
    }

    if (!last) {
        __syncthreads();
        // numer[o][c] = sum_p Q[o][p] * x[p][c]  (180 threads)
        if (t < 9 * CHN) {
            const int o = t / CHN, c = t % CHN;
            float accn = 0.f;
            #pragma unroll 8
            for (int p2 = 0; p2 < PIX; ++p2)
                accn += dq_lds[o * XSTR + p2] * x_lds[c * XSTR + p2];
            atomicAdd(&numer[(size_t)(b * SEG + sidx[o]) * CHN + c], accn);
        }
        if (t < 9) {
            float accd = 0.f;
            #pragma unroll 8
            for (int p2 = 0; p2 < PIX; ++p2)
                accd += dq_lds[t * XSTR + p2];
            atomicAdd(&denom[b * SEG + sidx[t]], accd);
        }
    }
}

// ---------------------------------------------------------------------------
// seed_update: seeds = numer/(denom+eps); recompute ss; reset accumulators
// one block (32 thr) per (b, s)
// ---------------------------------------------------------------------------
__global__ __launch_bounds__(32)
void seed_update_kernel(float* __restrict__ seeds,
                        float* __restrict__ numer,
                        float* __restrict__ denom,
                        float* __restrict__ ss)
{
    __shared__ float sqs[CHN];
    const int blk = blockIdx.x;
    const int b = blk >> 10;
    const int s = blk & (SEG - 1);
    const int t = threadIdx.x;

    const float d = denom[b * SEG + s];
    if (t < CHN) {
        size_t i = (size_t)(b * SEG + s) * CHN + t;
        float v = numer[i] / (d + EPSV);
        seeds[i] = v;
        numer[i] = 0.f;
        sqs[t] = v * v;
    }
    __syncthreads();
    if (t == 0) {
        float sq = 0.f;
        #pragma unroll
        for (int c = 0; c < CHN; ++c) sq += sqs[c];
        ss[b * SEG + s] = sq;
        denom[b * SEG + s] = 0.f;
    }
}

// ---------------------------------------------------------------------------
extern "C" void kernel_launch(void* const* d_in, const int* in_sizes, int n_in,
                              void* d_out, int out_size, void* d_ws, size_t ws_size,
                              hipStream_t stream)
{
    (void)in_sizes; (void)n_in; (void)out_size; (void)ws_size;
    const float* x = (const float*)d_in[0];
    float* out = (float*)d_out;

    char* ws = (char*)d_ws;
    float* seeds = (float*)(ws);                 // 8*1024*20*4 = 655360 B
    float* numer = (float*)(ws + 0x0A0000);      // 655360 B
    float* denom = (float*)(ws + 0x140000);      // 32768 B (contiguous after numer)
    float* ssbuf = (float*)(ws + 0x148000);      // 32768 B

    const int nblk = BATCH * SEG;                // 8192 cells

    seed_init_kernel<<<nblk, 256, 0, stream>>>(x, seeds, ssbuf);

    const int nzero = BATCH * SEG * CHN + BATCH * SEG;  // numer + denom
    zero_kernel<<<(nzero + 255) / 256, 256, 0, stream>>>(numer, nzero);

    for (int it = 0; it < 4; ++it) {
        cell_kernel<<<nblk, 256, 0, stream>>>(x, seeds, ssbuf, numer, denom,
                                              out, (it == 3) ? 1 : 0);
        if (it < 3)
            seed_update_kernel<<<nblk, 32, 0, stream>>>(seeds, numer, denom, ssbuf);
    }
}